// AdversarialLoss_64183991272155
// MI455X (gfx1250) — compile-verified
//
#include <hip/hip_runtime.h>
#include <hip/hip_bf16.h>

typedef float v2f __attribute__((ext_vector_type(2)));
typedef float v4f __attribute__((ext_vector_type(4)));
typedef float v8f __attribute__((ext_vector_type(8)));

#define VDIM 32000
#define ROWS_PER_BLOCK 16      // = WMMA M
#define WAVES 16               // 512 threads / block
#define TILE_W 32              // columns of the 16-row tile staged per wave iteration
#define NTILES (VDIM / TILE_W) // 1000
#define LDS_STRIDE 36          // padded row stride (floats) -> conflict-free A reads

// -ln(2)/V : folds the log2 -> ln conversion and the mean + minus sign
#define NEG_LN2_OVER_V (-0.69314718055994531f / (float)VDIM)

__global__ __launch_bounds__(WAVES * 32)
void AdversarialLoss_nll_rows_kernel(const float* __restrict__ pred,
                                     const int* __restrict__ target,
                                     float* __restrict__ out)
{
    __shared__ float tileLds[WAVES][ROWS_PER_BLOCK * LDS_STRIDE];
    __shared__ float redbuf[WAVES][ROWS_PER_BLOCK];

    const int tid  = threadIdx.x;
    const int wave = tid >> 5;
    const int lane = tid & 31;
    const int r0   = blockIdx.x * ROWS_PER_BLOCK;

    float* lds = tileLds[wave];

    // B = all-ones 4x16 matrix -> D[m][n] = sum_k A[m][k]
    v2f bones; bones.x = 1.0f; bones.y = 1.0f;
    v8f acc = {}; // C/D accumulator: per-row partial log2-sums (replicated over N)

    // Global staging pattern: per iteration i, 4 rows; 8 lanes x float4 = 128B
    // contiguous per row.
    const int g_row_off = lane >> 3;        // 0..3
    const int g_col     = (lane & 7) * 4;   // 0..28

    // A-fragment LDS read pattern (16x4 f32 A layout):
    // lanes 0-15: M=lane, K=kt*4+{0,1}; lanes 16-31: M=lane-16, K=kt*4+{2,3}
    const int a_row  = lane & 15;
    const int a_koff = 2 * (lane >> 4);     // 0 or 2

    for (int ct = wave; ct < NTILES; ct += WAVES) {
        const int c0 = ct * TILE_W;
        // Stage: load 16x32 tile (NT, read-once stream), raw v_log_f32 (log2),
        // write LDS. Inputs are in (1e-3, 1.001): no denorm/inf handling needed.
        #pragma unroll
        for (int i = 0; i < 4; ++i) {
            const int row = 4 * i + g_row_off;
            const float* p = pred + (size_t)(r0 + row) * VDIM + (c0 + g_col);
            v4f x = __builtin_nontemporal_load((const v4f*)p);
            v4f l;
            l.x = __builtin_amdgcn_logf(x.x);
            l.y = __builtin_amdgcn_logf(x.y);
            l.z = __builtin_amdgcn_logf(x.z);
            l.w = __builtin_amdgcn_logf(x.w);
            *(v4f*)&lds[row * LDS_STRIDE + g_col] = l;
        }
        // Consume: 8 chained accumulating WMMAs cover K = c0 .. c0+31.
        // (same-wave LDS store->load is in-order; compiler inserts s_wait_dscnt)
        #pragma unroll
        for (int kt = 0; kt < TILE_W / 4; ++kt) {
            v2f a = *(const v2f*)&lds[a_row * LDS_STRIDE + 4 * kt + a_koff];
            acc = __builtin_amdgcn_wmma_f32_16x16x4_f32(
                false, a, false, bones, (short)0, acc, false, false);
        }
    }

    // D layout: lane 0 VGPR m -> row m (m=0..7); lane 16 VGPR m -> row m+8.
    if ((lane & 15) == 0) {
        const int half = lane >> 4;
        #pragma unroll
        for (int m = 0; m < 8; ++m)
            redbuf[wave][half * 8 + m] = acc[m];
    }
    __syncthreads();

    // Final cross-wave reduce + target-class correction + fused ln2/mean scale.
    if (tid < ROWS_PER_BLOCK) {
        float s = 0.0f;
        #pragma unroll
        for (int w = 0; w < WAVES; ++w) s += redbuf[w][tid];
        const int row = r0 + tid;
        const int t = target[row];
        s -= __builtin_amdgcn_logf(pred[(size_t)row * VDIM + t]);
        out[row] = s * NEG_LN2_OVER_V;
    }
}

extern "C" void kernel_launch(void* const* d_in, const int* in_sizes, int n_in,
                              void* d_out, int out_size, void* d_ws, size_t ws_size,
                              hipStream_t stream) {
    const float* pred  = (const float*)d_in[0];
    const int* target  = (const int*)d_in[1];
    float* out         = (float*)d_out;

    const int B = in_sizes[1];                 // 8192 rows
    dim3 grid(B / ROWS_PER_BLOCK);             // 512 blocks of 16 rows
    dim3 block(WAVES * 32);                    // 512 threads = 16 wave32
    AdversarialLoss_nll_rows_kernel<<<grid, block, 0, stream>>>(pred, target, out);
}